// NCutsLoss_26027501813991
// MI455X (gfx1250) — compile-verified
//
#include <hip/hip_runtime.h>
#include <stdint.h>

// ---------------- problem constants (fixed by the reference) ----------------
#define WIN 9                       // 2*(RADIUS-1)+1
static constexpr int   N_   = 8;
static constexpr int   K_   = 4;
static constexpr int   H_   = 224;
static constexpr int   W_   = 224;
static constexpr int   HW_  = H_ * W_;            // 50176
static constexpr int   PW_  = W_ + 8;             // 232 (pad = RADIUS-1 = 4)
static constexpr int   PHW_ = PW_ * PW_;          // 53824
static constexpr float K_CONST = 4.0f;

static constexpr int THREADS     = 64;                     // 2 waves (wave32)
static constexpr int PIX_PER_THR = 2;                      // even-w pixel pair
static constexpr int PIX_PER_BLK = THREADS * PIX_PER_THR;  // 128
static constexpr int BLK_PER_IMG = HW_ / PIX_PER_BLK;      // 392
static constexpr int WTILE_DW    = PIX_PER_BLK * WIN*WIN;  // 10368 dwords = 41472 B

// -------- CDNA5 async global->LDS copy (ASYNCcnt-tracked) helpers ----------
__device__ __forceinline__ void async_lds_b128(uint32_t lds_byte, uint32_t gbyte,
                                               const void* sbase) {
  asm volatile("global_load_async_to_lds_b128 %0, %1, %2"
               :: "v"(lds_byte), "v"(gbyte), "s"(sbase) : "memory");
}
__device__ __forceinline__ void async_lds_b32(uint32_t lds_byte, uint32_t gbyte,
                                              const void* sbase) {
  asm volatile("global_load_async_to_lds_b32 %0, %1, %2"
               :: "v"(lds_byte), "v"(gbyte), "s"(sbase) : "memory");
}
__device__ __forceinline__ void wait_asynccnt0() {
  asm volatile("s_wait_asynccnt 0" ::: "memory");
}

__device__ __forceinline__ float wave32_sum(float v) {
  #pragma unroll
  for (int off = 16; off > 0; off >>= 1) v += __shfl_xor(v, off, 32);
  return v;
}

// --------------------------- main kernel -----------------------------------
// One block = 128 consecutive pixels (64 threads x 2-pixel pairs) of one
// image. The HBM-bound weight tile is staged into LDS with async copies;
// padded_seg rows are fetched as aligned float2 segments shared by the pair.
__global__ __launch_bounds__(THREADS)
void ncuts_main(const float* __restrict__ seg,
                const float* __restrict__ pseg,
                const float* __restrict__ weight,
                const float* __restrict__ sumw,
                float* __restrict__ partials) {
  __shared__ float wsm[WTILE_DW];          // mirrored copy of weight tile
  __shared__ float rsm[THREADS / 32][8];   // per-wave partials

  const int tid = threadIdx.x;
  const int b   = blockIdx.x;
  const int n   = b / BLK_PER_IMG;
  const int p0  = (b % BLK_PER_IMG) * PIX_PER_BLK;
  const int p   = p0 + 2 * tid;            // even; pair (p, p+1) shares a row

  // -- issue async weight staging first (coalesced: lane i <-> chunk i) -----
  const float* wbase = weight + (size_t)(n * HW_ + p0) * (WIN * WIN);
  const uint32_t lds0 = (uint32_t)(uintptr_t)(&wsm[0]);   // LDS byte offset
  #pragma unroll
  for (int i = 0; i < 40; ++i) {                 // 40 * 64 float4s = 10240 dw
    uint32_t byteoff = (uint32_t)(i * THREADS + tid) * 16u;
    async_lds_b128(lds0 + byteoff, byteoff, wbase);
  }
  #pragma unroll
  for (int i = 0; i < 2; ++i) {                  // tail: 128 dwords
    uint32_t byteoff = (uint32_t)(10240 + i * THREADS + tid) * 4u;
    async_lds_b32(lds0 + byteoff, byteoff, wbase);
  }

  // -- overlap: per-pair seg / sum_weight gathers while DMA is in flight ---
  const int h = p / W_, w = p % W_;               // w even
  float s0[K_], s1[K_];
  #pragma unroll
  for (int k = 0; k < K_; ++k) {
    float2 sv = *(const float2*)(seg + (size_t)(n * K_ + k) * HW_ + p);
    s0[k] = sv.x; s1[k] = sv.y;
  }
  const float2 swv = *(const float2*)(sumw + (size_t)n * HW_ + p);

  wait_asynccnt0();        // my wave's async copies landed in LDS
  __syncthreads();         // everyone's copies visible block-wide

  // -- 9x9 correlation for the pixel pair ----------------------------------
  // acc{0,1}[k] = sum_{m,j} pseg[k, h+m, w{+0,+1}+j] * weight{0,1}[m,j]
  const float* q0    = &wsm[(2 * tid)     * (WIN * WIN)];
  const float* q1    = &wsm[(2 * tid + 1) * (WIN * WIN)];
  const float* pbase = pseg + (size_t)n * K_ * PHW_;
  float acc0[K_] = {0.f, 0.f, 0.f, 0.f};
  float acc1[K_] = {0.f, 0.f, 0.f, 0.f};
  #pragma unroll 1
  for (int m = 0; m < WIN; ++m) {
    float w0[WIN], w1[WIN];
    #pragma unroll
    for (int j = 0; j < WIN; ++j) { w0[j] = q0[m * WIN + j]; w1[j] = q1[m * WIN + j]; }
    #pragma unroll
    for (int k = 0; k < K_; ++k) {
      // 10 floats cover both pixels' 9 taps; base is 8B-aligned (even dword)
      const float2* rowv = (const float2*)(pbase + (size_t)k * PHW_ +
                                           (size_t)(h + m) * PW_ + w);
      float f[10];
      #pragma unroll
      for (int t = 0; t < 5; ++t) { float2 e = rowv[t]; f[2*t] = e.x; f[2*t+1] = e.y; }
      #pragma unroll
      for (int j = 0; j < WIN; ++j) acc0[k] = fmaf(f[j],     w0[j], acc0[k]);
      #pragma unroll
      for (int j = 0; j < WIN; ++j) acc1[k] = fmaf(f[j + 1], w1[j], acc1[k]);
    }
  }

  // -- pair contributions: [A_0..A_3, V_0..V_3] ----------------------------
  float c8[8];
  #pragma unroll
  for (int k = 0; k < K_; ++k) {
    c8[k]     = acc0[k] * s0[k] + acc1[k] * s1[k];   // -> assocA[n,k]
    c8[4 + k] = swv.x   * s0[k] + swv.y   * s1[k];   // -> assocV[n,k]
  }

  // -- deterministic block reduction: wave32 trees, then cross-wave in LDS --
  #pragma unroll
  for (int c = 0; c < 8; ++c) c8[c] = wave32_sum(c8[c]);
  const int wave = tid >> 5, lane = tid & 31;
  if (lane == 0) {
    #pragma unroll
    for (int c = 0; c < 8; ++c) rsm[wave][c] = c8[c];
  }
  __syncthreads();
  if (tid < 8) {
    float s = 0.f;
    #pragma unroll
    for (int wv2 = 0; wv2 < THREADS / 32; ++wv2) s += rsm[wv2][tid];
    partials[(size_t)b * 8 + tid] = s;
  }
}

// ------------------------- finalize kernel ---------------------------------
// 8 blocks (one per image), 8 waves each: wave c reduces component c over the
// 392 per-block partials; thread 0 combines A/V and writes out[n].
__global__ __launch_bounds__(256)
void ncuts_finalize(const float* __restrict__ partials, float* __restrict__ out) {
  __shared__ float comp[8];
  const int n    = blockIdx.x;
  const int tid  = threadIdx.x;
  const int c    = tid >> 5;          // component handled by this wave
  const int lane = tid & 31;
  float s = 0.f;
  for (int i = lane; i < BLK_PER_IMG; i += 32)
    s += partials[((size_t)n * BLK_PER_IMG + i) * 8 + c];
  s = wave32_sum(s);
  if (lane == 0) comp[c] = s;
  __syncthreads();
  if (tid == 0) {
    float assoc = 0.f;
    #pragma unroll
    for (int k = 0; k < K_; ++k) assoc += comp[k] / comp[4 + k];
    out[n] = K_CONST - assoc;
  }
}

// ------------------------------ launcher -----------------------------------
extern "C" void kernel_launch(void* const* d_in, const int* in_sizes, int n_in,
                              void* d_out, int out_size, void* d_ws, size_t ws_size,
                              hipStream_t stream) {
  const float* seg  = (const float*)d_in[0];   // [8,4,224,224]
  const float* pseg = (const float*)d_in[1];   // [8,4,232,232]
  const float* wgt  = (const float*)d_in[2];   // [8,1,224,224,9,9]
  const float* sw   = (const float*)d_in[3];   // [8,1,224,224]
  float* out      = (float*)d_out;             // [8]
  float* partials = (float*)d_ws;              // N*392*8 floats = 100,352 B

  (void)in_sizes; (void)n_in; (void)out_size; (void)ws_size;

  ncuts_main<<<dim3(N_ * BLK_PER_IMG), dim3(THREADS), 0, stream>>>(
      seg, pseg, wgt, sw, partials);
  ncuts_finalize<<<dim3(N_), dim3(256), 0, stream>>>(partials, out);
}